// baseline2d_19224273617299
// MI455X (gfx1250) — compile-verified
//
#include <hip/hip_runtime.h>

typedef __attribute__((ext_vector_type(2))) float v2f;
typedef __attribute__((ext_vector_type(8))) float v8f;

#define BATCH 4
#define LSEQ  512
#define EDIM  1024
#define HDIM  128
#define TI 32
#define TJ 32
#define TP 33              // t-grid per tile (32 outputs + 1 halo row/col)
#define RPAD (HDIM + 4)    // break 64-bank stride-128 conflicts

// ---------------------------------------------------------------------------
// Kernel 1: hid = relu(A * W1 + b1) via V_WMMA_F32_16X16X4_F32 (fp32 WMMA).
// One wave computes one 16x16 tile of the (4096 x 128) fused rec|lig output.
// A layout (ISA 7.12.2, 32-bit A 16x4): lane<16 -> M=lane, K=vgpr (0,1);
// lane>=16 -> M=lane-16, K=vgpr+2.  B/C/D: N striped across lanes.
// ---------------------------------------------------------------------------
__global__ __launch_bounds__(128) void gemm_relu_wmma(
    const float* __restrict__ rec, const float* __restrict__ lig,
    const float* __restrict__ W1,  const float* __restrict__ b1,
    float* __restrict__ hidR, float* __restrict__ hidL)
{
    const int lane = threadIdx.x & 31;
    const int wave = threadIdx.x >> 5;
    const int tile = blockIdx.x * 4 + wave;      // 2048 tiles total
    const int n0 = (tile & 7) * 16;              // 8 n-tiles over H=128
    const int m0 = (tile >> 3) * 16;             // 256 m-tiles over 4096 rows

    const float* A;
    float* D;
    int row0;
    if (m0 < BATCH * LSEQ) { A = rec; D = hidR; row0 = m0; }
    else                   { A = lig; D = hidL; row0 = m0 - BATCH * LSEQ; }

    const int lo = lane & 15;
    const int hi = lane >> 4;
    const float* arow = A  + (size_t)(row0 + lo) * EDIM + 2 * hi;   // 8B aligned
    const float* bcol = W1 + (size_t)(2 * hi) * HDIM + (n0 + lo);

    v8f c = {};
    for (int k0 = 0; k0 < EDIM; k0 += 4) {
        v2f a = *(const v2f*)(arow + k0);                 // A[m][k0+2hi .. +1]
        v2f b;
        b.x = bcol[(size_t)k0 * HDIM];                    // B[k0+2hi  ][n]
        b.y = bcol[(size_t)(k0 + 1) * HDIM];              // B[k0+2hi+1][n]
        c = __builtin_amdgcn_wmma_f32_16x16x4_f32(
                false, a, false, b, (short)0, c, false, false);
    }

    const float bias = b1[n0 + lo];
    #pragma unroll
    for (int v = 0; v < 8; ++v) {                         // D: M = v + 8*hi
        float val = c[v] + bias;
        val = val > 0.f ? val : 0.f;                      // ReLU
        D[(size_t)(row0 + v + 8 * hi) * HDIM + (n0 + lo)] = val;
    }
}

// tanh(x) = 1 - 2/(exp2(x*2*log2 e)+1); inputs are >= 0 (ReLU*ReLU), safe.
__device__ __forceinline__ float fast_tanh(float x) {
    float e = __builtin_amdgcn_exp2f(x * 2.8853900817779268f);
    return 1.0f - 2.0f * __builtin_amdgcn_rcpf(e + 1.0f);
}

// ---------------------------------------------------------------------------
// Kernel 2: fused tanh-outer-product + 2x2 conv + max, no 537MB intermediate.
// Block = one 32x32 tile of y for one batch; t computed on 33x33 grid in LDS.
// ---------------------------------------------------------------------------
__global__ __launch_bounds__(256) void bilinear_tanh_conv_max(
    const float* __restrict__ hidR, const float* __restrict__ hidL,
    const float* __restrict__ convW, const float* __restrict__ convB,
    float* __restrict__ partial)
{
    __shared__ float Rs[TP][RPAD];
    __shared__ float Ls[TP][RPAD];
    __shared__ float Wq[4][HDIM];
    __shared__ float Ts[4][TP * TP];
    __shared__ float red[256];

    const int b   = blockIdx.y;
    const int i0  = (blockIdx.x >> 4) * TI;
    const int j0  = (blockIdx.x & 15) * TJ;
    const int tid = threadIdx.x;

    // Stage hid tiles with the -1 halo row/col (zero padding => tanh(0)=0).
    for (int idx = tid; idx < TP * HDIM; idx += 256) {
        int r = idx / HDIM, h = idx - r * HDIM;
        int gi = i0 - 1 + r;
        int gj = j0 - 1 + r;
        Rs[r][h] = (gi >= 0) ? hidR[(size_t)(b * LSEQ + gi) * HDIM + h] : 0.f;
        Ls[r][h] = (gj >= 0) ? hidL[(size_t)(b * LSEQ + gj) * HDIM + h] : 0.f;
    }
    for (int idx = tid; idx < 4 * HDIM; idx += 256) {
        int h = idx >> 2, q = idx & 3;
        Wq[q][h] = convW[h * 4 + q];           // conv_w is (1,H,2,2)
    }
    __syncthreads();

    // t_q(i,j) = sum_h w[h,q] * tanh(r[i,h]*l[j,h]) over the 33x33 grid.
    for (int p = tid; p < TP * TP; p += 256) {
        const int il = p / TP, jl = p - il * TP;
        const float* rr = &Rs[il][0];
        const float* ll = &Ls[jl][0];
        float a0 = 0.f, a1 = 0.f, a2 = 0.f, a3 = 0.f;
        #pragma unroll 8
        for (int h = 0; h < HDIM; ++h) {
            float v = fast_tanh(rr[h] * ll[h]);
            a0 += Wq[0][h] * v;
            a1 += Wq[1][h] * v;
            a2 += Wq[2][h] * v;
            a3 += Wq[3][h] * v;
        }
        Ts[0][p] = a0; Ts[1][p] = a1; Ts[2][p] = a2; Ts[3][p] = a3;
    }
    __syncthreads();

    // y[i',j'] = cb + t00(i'-1,j'-1) + t01(i'-1,j') + t10(i',j'-1) + t11(i',j')
    const float cb = convB[0];
    float mx = -3.402823466e38f;
    for (int q = tid; q < TI * TJ; q += 256) {
        int li = (q >> 5) + 1, lj = (q & 31) + 1;
        float y = cb + Ts[0][(li - 1) * TP + (lj - 1)]
                     + Ts[1][(li - 1) * TP + lj]
                     + Ts[2][li * TP + (lj - 1)]
                     + Ts[3][li * TP + lj];
        mx = fmaxf(mx, y);
    }
    red[tid] = mx;
    __syncthreads();
    for (int s = 128; s > 0; s >>= 1) {
        if (tid < s) red[tid] = fmaxf(red[tid], red[tid + s]);
        __syncthreads();
    }
    if (tid == 0) partial[b * 256 + blockIdx.x] = red[0];
}

// ---------------------------------------------------------------------------
// Kernel 3: per-batch max over 256 tile partials, then sigmoid.
// ---------------------------------------------------------------------------
__global__ __launch_bounds__(256) void reduce_sigmoid(
    const float* __restrict__ partial, float* __restrict__ out)
{
    __shared__ float red[256];
    const int b = blockIdx.x;
    red[threadIdx.x] = partial[b * 256 + threadIdx.x];
    __syncthreads();
    for (int s = 128; s > 0; s >>= 1) {
        if (threadIdx.x < s)
            red[threadIdx.x] = fmaxf(red[threadIdx.x], red[threadIdx.x + s]);
        __syncthreads();
    }
    if (threadIdx.x == 0) {
        float m = red[0];
        float e = __builtin_amdgcn_exp2f(-m * 1.4426950408889634f);
        out[b] = __builtin_amdgcn_rcpf(1.0f + e);
    }
}

extern "C" void kernel_launch(void* const* d_in, const int* in_sizes, int n_in,
                              void* d_out, int out_size, void* d_ws, size_t ws_size,
                              hipStream_t stream) {
    (void)in_sizes; (void)n_in; (void)out_size; (void)ws_size;
    const float* rec = (const float*)d_in[0];
    const float* lig = (const float*)d_in[1];
    const float* W1  = (const float*)d_in[2];
    const float* b1  = (const float*)d_in[3];
    const float* cw  = (const float*)d_in[4];
    const float* cb  = (const float*)d_in[5];

    float* hidR    = (float*)d_ws;                         // 4*512*128 fp32
    float* hidL    = hidR + (size_t)BATCH * LSEQ * HDIM;   // 4*512*128 fp32
    float* partial = hidL + (size_t)BATCH * LSEQ * HDIM;   // 4*256 fp32

    gemm_relu_wmma<<<512, 128, 0, stream>>>(rec, lig, W1, b1, hidR, hidL);

    dim3 g2(256, BATCH);
    bilinear_tanh_conv_max<<<g2, 256, 0, stream>>>(hidR, hidL, cw, cb, partial);

    reduce_sigmoid<<<BATCH, 256, 0, stream>>>(partial, (float*)d_out);
}